// GraphToLatent_23175643529964
// MI455X (gfx1250) — compile-verified
//
#include <hip/hip_runtime.h>

// GINE GNN -> latent tokens, CDNA5 (gfx1250) wave32 + V_WMMA_F32_16X16X4_F32.
// All GEMMs use the fp32 WMMA path (exact-precision vs fp32 reference); the
// problem is L2-gather/atomic bound (h/aggr ~51MB each, fit in 192MB L2).

typedef float v2f_t __attribute__((ext_vector_type(2)));
typedef float v8f_t __attribute__((ext_vector_type(8)));

#define HDIM 256
#define LSR  260   // LDS row stride (floats); pad 4 -> conflict-free b64 frag loads

static __device__ __forceinline__ v8f_t wmma_f32(v2f_t a, v2f_t b, v8f_t c) {
  // D(16x16,f32) = A(16x4,f32) x B(4x16,f32) + C
  return __builtin_amdgcn_wmma_f32_16x16x4_f32(false, a, false, b, (short)0, c,
                                               false, false);
}

// ---------------------------------------------------------------------------
// Pack B [K,N] row-major f32 into fragment-major layout:
// Bp[((ct*(K/4)+ks)*32+lane)*2 + {0,1}] = B[ks*4 + 2*(lane>>4) + {0,1}][ct*16 + (lane&15)]
// ---------------------------------------------------------------------------
__global__ __launch_bounds__(256) void pack_b_kernel(const float* __restrict__ B,
                                                     float* __restrict__ Bp,
                                                     int K, int N) {
  int t = blockIdx.x * 256 + threadIdx.x;
  int total = (K >> 2) * (N >> 4) * 32;
  if (t >= total) return;
  int lane = t & 31;
  int rest = t >> 5;
  int ks = rest % (K >> 2);
  int ct = rest / (K >> 2);
  int k = ks * 4 + 2 * (lane >> 4);
  int n = ct * 16 + (lane & 15);
  Bp[(size_t)t * 2 + 0] = B[(size_t)k * N + n];
  Bp[(size_t)t * 2 + 1] = B[(size_t)(k + 1) * N + n];
}

// ---------------------------------------------------------------------------
// h = x @ node_w + node_b   (K=9, scalar FMA; trivial cost)
// ---------------------------------------------------------------------------
__global__ __launch_bounds__(256) void node_embed_kernel(
    const float* __restrict__ x, const float* __restrict__ nw,
    const float* __restrict__ nb, float* __restrict__ h, int nNodes) {
  int t = blockIdx.x * 256 + threadIdx.x;
  if (t >= nNodes * 64) return;
  int n = t >> 6;
  int c4 = (t & 63) << 2;
  float4 acc = *(const float4*)(nb + c4);
#pragma unroll
  for (int j = 0; j < 9; ++j) {
    float xv = x[(size_t)n * 9 + j];
    float4 w = *(const float4*)(nw + j * HDIM + c4);
    acc.x += xv * w.x; acc.y += xv * w.y; acc.z += xv * w.z; acc.w += xv * w.w;
  }
  *(float4*)(h + (size_t)n * HDIM + c4) = acc;
}

// ---------------------------------------------------------------------------
// Edge message + scatter: aggr[dst] += relu(h[src] + edge_attr@ew + eb)
// ---------------------------------------------------------------------------
__global__ __launch_bounds__(256) void edge_kernel(
    const float* __restrict__ h, const int* __restrict__ ei,
    const float* __restrict__ ea, const float* __restrict__ ew,
    const float* __restrict__ eb, float* __restrict__ aggr, int nEdges) {
  long t = (long)blockIdx.x * 256 + threadIdx.x;
  if (t >= (long)nEdges * 64) return;
  int e = (int)(t >> 6);
  int c4 = ((int)t & 63) << 2;
  int src = ei[e];
  int dst = ei[nEdges + e];
  float a0 = ea[(size_t)e * 3 + 0];
  float a1 = ea[(size_t)e * 3 + 1];
  float a2 = ea[(size_t)e * 3 + 2];
  float4 w0 = *(const float4*)(ew + 0 * HDIM + c4);
  float4 w1 = *(const float4*)(ew + 1 * HDIM + c4);
  float4 w2 = *(const float4*)(ew + 2 * HDIM + c4);
  float4 bv = *(const float4*)(eb + c4);
  float4 hv = *(const float4*)(h + (size_t)src * HDIM + c4);
  float mx = fmaxf(hv.x + a0 * w0.x + a1 * w1.x + a2 * w2.x + bv.x, 0.f);
  float my = fmaxf(hv.y + a0 * w0.y + a1 * w1.y + a2 * w2.y + bv.y, 0.f);
  float mz = fmaxf(hv.z + a0 * w0.z + a1 * w1.z + a2 * w2.z + bv.z, 0.f);
  float mw = fmaxf(hv.w + a0 * w0.w + a1 * w1.w + a2 * w2.w + bv.w, 0.f);
  float* ag = aggr + (size_t)dst * HDIM + c4;
  atomicAdd(ag + 0, mx);
  atomicAdd(ag + 1, my);
  atomicAdd(ag + 2, mz);
  atomicAdd(ag + 3, mw);
}

// ---------------------------------------------------------------------------
// Fused per-layer node update:
//   z = (1+eps)h + aggr;  y = relu(z@w1 + b1);  o = y@w2 + b2;
//   h = LayerNorm(h + o) * g + b
// Block: 256 threads = 8 waves; 64-row strip; single LDS buffer (dynamic).
// Each wave computes 4 row-tiles x 2 col-tiles via WMMA f32 16x16x4.
// ---------------------------------------------------------------------------
__global__ __launch_bounds__(256) void mlp_ln_kernel(
    float* __restrict__ h, const float* __restrict__ aggr,
    const float* __restrict__ w1p, const float* __restrict__ b1,
    const float* __restrict__ w2p, const float* __restrict__ b2,
    const float* __restrict__ epsArr, int layer,
    const float* __restrict__ lnG, const float* __restrict__ lnB, int nRows) {
  extern __shared__ float smem[];
  float* sA = smem;                 // 64 * LSR
  float* sMu = smem + 64 * LSR;     // 64
  float* sRs = sMu + 64;            // 64

  const int tid = threadIdx.x;
  const int row0 = blockIdx.x * 64;
  const int wave = tid >> 5;
  const int lane = tid & 31;
  const int laneLo = lane & 15;
  const int laneHi = lane >> 4;
  const int ct0 = wave * 2;
  const float epsv = 1.0f + epsArr[layer];

  // phase 0: z -> sA
  for (int i = tid; i < 64 * 64; i += 256) {
    int r = i >> 6, c4 = (i & 63) << 2;
    int gr = row0 + r;
    float4 z = make_float4(0.f, 0.f, 0.f, 0.f);
    if (gr < nRows) {
      float4 hv = *(const float4*)(h + (size_t)gr * HDIM + c4);
      float4 av = *(const float4*)(aggr + (size_t)gr * HDIM + c4);
      z.x = epsv * hv.x + av.x; z.y = epsv * hv.y + av.y;
      z.z = epsv * hv.z + av.z; z.w = epsv * hv.w + av.w;
    }
    *(float4*)(&sA[r * LSR + c4]) = z;
  }
  __syncthreads();

  // GEMM1: acc = z @ w1
  v8f_t acc[4][2] = {};
  {
    const float* bp0 = w1p + ((size_t)(ct0 + 0) * 64 * 32 + lane) * 2;
    const float* bp1 = w1p + ((size_t)(ct0 + 1) * 64 * 32 + lane) * 2;
#pragma unroll 4
    for (int ks = 0; ks < 64; ++ks) {
      int kk = ks * 4 + 2 * laneHi;
      v2f_t a0 = *(const v2f_t*)(&sA[(0 * 16 + laneLo) * LSR + kk]);
      v2f_t a1 = *(const v2f_t*)(&sA[(1 * 16 + laneLo) * LSR + kk]);
      v2f_t a2 = *(const v2f_t*)(&sA[(2 * 16 + laneLo) * LSR + kk]);
      v2f_t a3 = *(const v2f_t*)(&sA[(3 * 16 + laneLo) * LSR + kk]);
      v2f_t bf0 = *(const v2f_t*)(bp0 + ks * 64);
      v2f_t bf1 = *(const v2f_t*)(bp1 + ks * 64);
      acc[0][0] = wmma_f32(a0, bf0, acc[0][0]);
      acc[1][0] = wmma_f32(a1, bf0, acc[1][0]);
      acc[2][0] = wmma_f32(a2, bf0, acc[2][0]);
      acc[3][0] = wmma_f32(a3, bf0, acc[3][0]);
      acc[0][1] = wmma_f32(a0, bf1, acc[0][1]);
      acc[1][1] = wmma_f32(a1, bf1, acc[1][1]);
      acc[2][1] = wmma_f32(a2, bf1, acc[2][1]);
      acc[3][1] = wmma_f32(a3, bf1, acc[3][1]);
    }
  }
  __syncthreads();  // all waves done reading z

  // y = relu(acc + b1) -> sA (overwrite)
#pragma unroll
  for (int ci = 0; ci < 2; ++ci) {
    int col = (ct0 + ci) * 16 + laneLo;
    float bv = b1[col];
#pragma unroll
    for (int rt = 0; rt < 4; ++rt)
#pragma unroll
      for (int i = 0; i < 8; ++i) {
        int r = rt * 16 + i + 8 * laneHi;
        sA[r * LSR + col] = fmaxf(acc[rt][ci][i] + bv, 0.f);
      }
  }
  __syncthreads();

  // GEMM2: acc2 = y @ w2
  v8f_t acc2[4][2] = {};
  {
    const float* bp0 = w2p + ((size_t)(ct0 + 0) * 64 * 32 + lane) * 2;
    const float* bp1 = w2p + ((size_t)(ct0 + 1) * 64 * 32 + lane) * 2;
#pragma unroll 4
    for (int ks = 0; ks < 64; ++ks) {
      int kk = ks * 4 + 2 * laneHi;
      v2f_t a0 = *(const v2f_t*)(&sA[(0 * 16 + laneLo) * LSR + kk]);
      v2f_t a1 = *(const v2f_t*)(&sA[(1 * 16 + laneLo) * LSR + kk]);
      v2f_t a2 = *(const v2f_t*)(&sA[(2 * 16 + laneLo) * LSR + kk]);
      v2f_t a3 = *(const v2f_t*)(&sA[(3 * 16 + laneLo) * LSR + kk]);
      v2f_t bf0 = *(const v2f_t*)(bp0 + ks * 64);
      v2f_t bf1 = *(const v2f_t*)(bp1 + ks * 64);
      acc2[0][0] = wmma_f32(a0, bf0, acc2[0][0]);
      acc2[1][0] = wmma_f32(a1, bf0, acc2[1][0]);
      acc2[2][0] = wmma_f32(a2, bf0, acc2[2][0]);
      acc2[3][0] = wmma_f32(a3, bf0, acc2[3][0]);
      acc2[0][1] = wmma_f32(a0, bf1, acc2[0][1]);
      acc2[1][1] = wmma_f32(a1, bf1, acc2[1][1]);
      acc2[2][1] = wmma_f32(a2, bf1, acc2[2][1]);
      acc2[3][1] = wmma_f32(a3, bf1, acc2[3][1]);
    }
  }
  __syncthreads();  // all waves done reading y

  // o + b2 + residual(h) -> sA
#pragma unroll
  for (int ci = 0; ci < 2; ++ci) {
    int col = (ct0 + ci) * 16 + laneLo;
    float bv = b2[col];
#pragma unroll
    for (int rt = 0; rt < 4; ++rt)
#pragma unroll
      for (int i = 0; i < 8; ++i) {
        int r = rt * 16 + i + 8 * laneHi;
        int gr = row0 + r;
        float res = (gr < nRows) ? h[(size_t)gr * HDIM + col] : 0.f;
        sA[r * LSR + col] = acc2[rt][ci][i] + bv + res;
      }
  }
  __syncthreads();

  // LayerNorm stats per row
  if (tid < 64) {
    float s = 0.f, s2 = 0.f;
    for (int c = 0; c < HDIM; ++c) {
      float v = sA[tid * LSR + c];
      s += v; s2 += v * v;
    }
    float mu = s * (1.0f / HDIM);
    float var = s2 * (1.0f / HDIM) - mu * mu;
    sMu[tid] = mu;
    sRs[tid] = rsqrtf(var + 1e-5f);
  }
  __syncthreads();

  const float* g = lnG + (size_t)layer * HDIM;
  const float* bb = lnB + (size_t)layer * HDIM;
  for (int i = tid; i < 64 * 64; i += 256) {
    int r = i >> 6, c4 = (i & 63) << 2;
    int gr = row0 + r;
    if (gr < nRows) {
      float mu = sMu[r], rs = sRs[r];
      float4 o;
      o.x = (sA[r * LSR + c4 + 0] - mu) * rs * g[c4 + 0] + bb[c4 + 0];
      o.y = (sA[r * LSR + c4 + 1] - mu) * rs * g[c4 + 1] + bb[c4 + 1];
      o.z = (sA[r * LSR + c4 + 2] - mu) * rs * g[c4 + 2] + bb[c4 + 2];
      o.w = (sA[r * LSR + c4 + 3] - mu) * rs * g[c4 + 3] + bb[c4 + 3];
      *(float4*)(h + (size_t)gr * HDIM + c4) = o;
    }
  }
}

// ---------------------------------------------------------------------------
// Generic WMMA GEMM: C[M,N] = act(A[M,K] @ Bpacked + bias); K chunked by 256.
// grid = (ceil(M/64), N/256); 256 threads.
// ---------------------------------------------------------------------------
__global__ __launch_bounds__(256) void gemm_kernel(
    const float* __restrict__ A, const float* __restrict__ Bp,
    const float* __restrict__ bias, float* __restrict__ C,
    int M, int K, int N, int doRelu) {
  extern __shared__ float smem[];
  float* sA = smem;  // 64 * LSR
  const int tid = threadIdx.x;
  const int wave = tid >> 5;
  const int lane = tid & 31;
  const int laneLo = lane & 15;
  const int laneHi = lane >> 4;
  const int row0 = blockIdx.x * 64;
  const int ctg0 = blockIdx.y * 16 + wave * 2;
  const int ksteps = K >> 2;

  v8f_t acc[4][2] = {};
  for (int kc = 0; kc < K; kc += 256) {
    __syncthreads();
    for (int i = tid; i < 64 * 64; i += 256) {
      int r = i >> 6, c4 = (i & 63) << 2;
      int gr = row0 + r;
      float4 v = make_float4(0.f, 0.f, 0.f, 0.f);
      if (gr < M) v = *(const float4*)(A + (size_t)gr * K + kc + c4);
      *(float4*)(&sA[r * LSR + c4]) = v;
    }
    __syncthreads();
    const float* bp0 = Bp + (((size_t)(ctg0 + 0) * ksteps + (kc >> 2)) * 32 + lane) * 2;
    const float* bp1 = Bp + (((size_t)(ctg0 + 1) * ksteps + (kc >> 2)) * 32 + lane) * 2;
#pragma unroll 4
    for (int ks = 0; ks < 64; ++ks) {
      int kk = ks * 4 + 2 * laneHi;
      v2f_t a0 = *(const v2f_t*)(&sA[(0 * 16 + laneLo) * LSR + kk]);
      v2f_t a1 = *(const v2f_t*)(&sA[(1 * 16 + laneLo) * LSR + kk]);
      v2f_t a2 = *(const v2f_t*)(&sA[(2 * 16 + laneLo) * LSR + kk]);
      v2f_t a3 = *(const v2f_t*)(&sA[(3 * 16 + laneLo) * LSR + kk]);
      v2f_t bf0 = *(const v2f_t*)(bp0 + ks * 64);
      v2f_t bf1 = *(const v2f_t*)(bp1 + ks * 64);
      acc[0][0] = wmma_f32(a0, bf0, acc[0][0]);
      acc[1][0] = wmma_f32(a1, bf0, acc[1][0]);
      acc[2][0] = wmma_f32(a2, bf0, acc[2][0]);
      acc[3][0] = wmma_f32(a3, bf0, acc[3][0]);
      acc[0][1] = wmma_f32(a0, bf1, acc[0][1]);
      acc[1][1] = wmma_f32(a1, bf1, acc[1][1]);
      acc[2][1] = wmma_f32(a2, bf1, acc[2][1]);
      acc[3][1] = wmma_f32(a3, bf1, acc[3][1]);
    }
  }

#pragma unroll
  for (int ci = 0; ci < 2; ++ci) {
    int col = (ctg0 + ci) * 16 + laneLo;
    float bv = bias[col];
#pragma unroll
    for (int rt = 0; rt < 4; ++rt)
#pragma unroll
      for (int i = 0; i < 8; ++i) {
        int gr = row0 + rt * 16 + i + 8 * laneHi;
        if (gr < M) {
          float v = acc[rt][ci][i] + bv;
          if (doRelu) v = fmaxf(v, 0.f);
          C[(size_t)gr * N + col] = v;
        }
      }
  }
}

// ---------------------------------------------------------------------------
// Mean pooling over graphs
// ---------------------------------------------------------------------------
__global__ __launch_bounds__(256) void pool_sum_kernel(
    const float* __restrict__ h, const int* __restrict__ batch,
    float* __restrict__ gsum, int nNodes) {
  int t = blockIdx.x * 256 + threadIdx.x;
  if (t >= nNodes * 64) return;
  int n = t >> 6;
  int c4 = (t & 63) << 2;
  int b = batch[n];
  float4 v = *(const float4*)(h + (size_t)n * HDIM + c4);
  float* gs = gsum + (size_t)b * HDIM + c4;
  atomicAdd(gs + 0, v.x);
  atomicAdd(gs + 1, v.y);
  atomicAdd(gs + 2, v.z);
  atomicAdd(gs + 3, v.w);
}

__global__ __launch_bounds__(256) void pool_cnt_kernel(
    const int* __restrict__ batch, float* __restrict__ cnt, int nNodes) {
  int t = blockIdx.x * 256 + threadIdx.x;
  if (t >= nNodes) return;
  atomicAdd(&cnt[batch[t]], 1.0f);
}

__global__ __launch_bounds__(256) void pool_div_kernel(
    float* __restrict__ gsum, const float* __restrict__ cnt) {
  int t = blockIdx.x * 256 + threadIdx.x;
  if (t >= 256 * HDIM) return;
  int b = t >> 8;
  gsum[t] = gsum[t] / fmaxf(cnt[b], 1.0f);
}

// ---------------------------------------------------------------------------
extern "C" void kernel_launch(void* const* d_in, const int* in_sizes, int n_in,
                              void* d_out, int out_size, void* d_ws,
                              size_t ws_size, hipStream_t stream) {
  const float* x      = (const float*)d_in[0];
  const int*   ei     = (const int*)d_in[1];
  const float* ea     = (const float*)d_in[2];
  const int*   batch  = (const int*)d_in[3];
  const float* node_w = (const float*)d_in[4];
  const float* node_b = (const float*)d_in[5];
  const float* edge_w = (const float*)d_in[6];
  const float* edge_b = (const float*)d_in[7];
  const float* w1     = (const float*)d_in[8];
  const float* b1     = (const float*)d_in[9];
  const float* w2     = (const float*)d_in[10];
  const float* b2     = (const float*)d_in[11];
  const float* epsArr = (const float*)d_in[12];
  const float* lnG    = (const float*)d_in[13];
  const float* lnB    = (const float*)d_in[14];
  const float* r_w1   = (const float*)d_in[15];
  const float* r_b1   = (const float*)d_in[16];
  const float* r_w2   = (const float*)d_in[17];
  const float* r_b2   = (const float*)d_in[18];
  float* out = (float*)d_out;

  const int N_ = in_sizes[0] / 9;    // 50000
  const int E_ = in_sizes[2] / 3;    // 300000
  const int L_ = 3;

  // workspace carve-out
  char* ws = (char*)d_ws;
  size_t off = 0;
  auto carve = [&](size_t bytes) -> void* {
    void* p = ws + off;
    off = (off + bytes + 255) & ~(size_t)255;
    return p;
  };
  float* h    = (float*)carve((size_t)N_ * HDIM * 4);
  float* aggr = (float*)carve((size_t)N_ * HDIM * 4);
  float* gsum = (float*)carve((size_t)256 * HDIM * 4);
  float* cnt  = (float*)carve(1024);
  float* a1   = (float*)carve((size_t)256 * 1024 * 4);
  float* w1p  = (float*)carve((size_t)L_ * HDIM * HDIM * 4);
  float* w2p  = (float*)carve((size_t)L_ * HDIM * HDIM * 4);
  float* rw1p = (float*)carve((size_t)HDIM * 1024 * 4);
  float* rw2p = (float*)carve((size_t)1024 * 4096 * 4);

  const size_t shMlp = (size_t)(64 * LSR + 128) * 4;  // 67072
  const size_t shGem = (size_t)(64 * LSR) * 4;        // 66560

  // ---- pack weights (every call; deterministic) ----
  {
    int tot = (HDIM >> 2) * (HDIM >> 4) * 32;  // 32768
    int blk = (tot + 255) / 256;
    for (int l = 0; l < L_; ++l) {
      pack_b_kernel<<<blk, 256, 0, stream>>>(w1 + (size_t)l * HDIM * HDIM,
                                             w1p + (size_t)l * HDIM * HDIM,
                                             HDIM, HDIM);
      pack_b_kernel<<<blk, 256, 0, stream>>>(w2 + (size_t)l * HDIM * HDIM,
                                             w2p + (size_t)l * HDIM * HDIM,
                                             HDIM, HDIM);
    }
    int tot1 = (HDIM >> 2) * (1024 >> 4) * 32;
    pack_b_kernel<<<(tot1 + 255) / 256, 256, 0, stream>>>(r_w1, rw1p, HDIM, 1024);
    int tot2 = (1024 >> 2) * (4096 >> 4) * 32;
    pack_b_kernel<<<(tot2 + 255) / 256, 256, 0, stream>>>(r_w2, rw2p, 1024, 4096);
  }

  // ---- node embedding ----
  node_embed_kernel<<<(N_ * 64 + 255) / 256, 256, 0, stream>>>(x, node_w,
                                                               node_b, h, N_);

  // ---- GINE layers ----
  const int mlpBlocks = (N_ + 63) / 64;
  for (int l = 0; l < L_; ++l) {
    hipMemsetAsync(aggr, 0, (size_t)N_ * HDIM * 4, stream);
    long etot = (long)E_ * 64;
    edge_kernel<<<(int)((etot + 255) / 256), 256, 0, stream>>>(
        h, ei, ea, edge_w + (size_t)l * 3 * HDIM, edge_b + (size_t)l * HDIM,
        aggr, E_);
    mlp_ln_kernel<<<mlpBlocks, 256, shMlp, stream>>>(
        h, aggr, w1p + (size_t)l * HDIM * HDIM, b1 + (size_t)l * HDIM,
        w2p + (size_t)l * HDIM * HDIM, b2 + (size_t)l * HDIM, epsArr, l, lnG,
        lnB, N_);
  }

  // ---- mean pool ----
  hipMemsetAsync(gsum, 0, (size_t)256 * HDIM * 4, stream);
  hipMemsetAsync(cnt, 0, 1024, stream);
  pool_sum_kernel<<<(N_ * 64 + 255) / 256, 256, 0, stream>>>(h, batch, gsum, N_);
  pool_cnt_kernel<<<(N_ + 255) / 256, 256, 0, stream>>>(batch, cnt, N_);
  pool_div_kernel<<<(256 * HDIM + 255) / 256, 256, 0, stream>>>(gsum, cnt);

  // ---- readout MLP: a1 = relu(g @ r_w1 + r_b1); out = a1 @ r_w2 + r_b2 ----
  {
    dim3 g1((256 + 63) / 64, 1024 / 256);
    gemm_kernel<<<g1, 256, shGem, stream>>>(gsum, rw1p, r_b1, a1, 256, HDIM,
                                            1024, 1);
    dim3 g2((256 + 63) / 64, 4096 / 256);
    gemm_kernel<<<g2, 256, shGem, stream>>>(a1, rw2p, r_b2, out, 256, 1024,
                                            4096, 0);
  }
}